// DetectionPostProcessV1_15719580304110
// MI455X (gfx1250) — compile-verified
//
#include <hip/hip_runtime.h>

// ---------------------------------------------------------------------------
// Detection post-process for MI455X (gfx1250, wave32).
// Memory-bound pipeline (~198 MB in / 23.3 TB/s ~= 8.5 us floor). No matmul
// structure -> WMMA not applicable; CDNA5 leverage = wave32 block sizing,
// async global->LDS staging (ASYNCcnt) and non-temporal streaming loads.
// ---------------------------------------------------------------------------

#define NB      8
#define NA      9
#define NC      80
#define CHN     (NA * (4 + NC))      /* 756 channels */
#define H0      80
#define W0      80
#define H1      40
#define W1      40
#define HW0     (H0 * W0)            /* 6400 */
#define HW1     (H1 * W1)            /* 1600 */
#define NCAND0  (HW0 * NA)           /* 57600 */
#define NCAND1  (HW1 * NA)           /* 14400 */
#define NCAND   (NCAND0 + NCAND1)    /* 72000 */
#define PRE     1000
#define POST    300
#define SCORE_THR 0.05f
#define NMS_THR   0.5f
#define IMG_WH    640.0f
#define BBOX_CLIP 4.1351665567423560f /* log(1000/16) */
#define TILE      2048u               /* score elements per async tile */

// ---------------- monotone float<->u32 key (descending select) -------------
__device__ __forceinline__ unsigned keyOf(float f) {
    unsigned u = __float_as_uint(f);
    return (u & 0x80000000u) ? ~u : (u | 0x80000000u);
}
__device__ __forceinline__ float keyToFloat(unsigned k) {
    unsigned u = (k & 0x80000000u) ? (k & 0x7fffffffu) : ~k;
    return __uint_as_float(u);
}

// ---------------- CDNA5 async global->LDS copy (ASYNCcnt) ------------------
__device__ __forceinline__ void asyncLoadB64(unsigned ldsByteAddr, const float* g) {
    unsigned long long ga = (unsigned long long)g;
    asm volatile("global_load_async_to_lds_b64 %0, %1, off"
                 :: "v"(ldsByteAddr), "v"(ga)
                 : "memory");
}
__device__ __forceinline__ void waitAsync0() {
    asm volatile("s_wait_asynccnt 0" ::: "memory");
}
__device__ __forceinline__ void waitAsync1() {
    asm volatile("s_wait_asynccnt 1" ::: "memory");
}

// Stream one image's 72000 scores through LDS with double-buffered async
// copies. Each lane consumes only the two floats it staged itself, so only
// per-wave s_wait_asynccnt ordering is required (no block barriers inside).
template <typename F>
__device__ __forceinline__ void streamScores(const float* __restrict__ sc,
                                             float tile[2][TILE],
                                             unsigned ldsT0, unsigned ldsT1,
                                             unsigned tid, F fn) {
    const unsigned numTiles = (NCAND + TILE - 1) / TILE;  // 36 (last partial)
    {
        unsigned e0 = tid * 2u;
        if (e0 < NCAND) asyncLoadB64(ldsT0 + tid * 8u, sc + e0);
    }
    for (unsigned t = 0; t < numTiles; ++t) {
        if (t + 1u < numTiles) {
            unsigned en    = (t + 1u) * TILE + tid * 2u;
            unsigned nAddr = ((t + 1u) & 1u) ? ldsT1 : ldsT0;
            if (en < NCAND) asyncLoadB64(nAddr + tid * 8u, sc + en);
            waitAsync1();   // in-order completion => tile t is resident
        } else {
            waitAsync0();
        }
        const float* tb = tile[t & 1u];
        unsigned base = t * TILE + tid * 2u;
#pragma unroll
        for (int k = 0; k < 2; ++k) {
            unsigned e = base + (unsigned)k;
            if (e < NCAND) fn(e, tb[tid * 2u + (unsigned)k]);
        }
    }
}

// ---------------------------------------------------------------------------
// Kernel A: per-anchor max class score. a-major / hw-minor thread mapping so
// every class-channel read is fully coalesced. NT loads: 184 MB one-shot
// stream should not evict L2 working data. Unroll 8 -> 8 loads in flight
// per wave between waits (loads are HW-strided, so widen via MLP not width).
// ---------------------------------------------------------------------------
__global__ void max_score_kernel(const float* __restrict__ d0,
                                 const float* __restrict__ d1,
                                 float* __restrict__ scores) {
    int flat = blockIdx.x * blockDim.x + threadIdx.x;
    if (flat >= NB * NCAND) return;
    int b = flat / NCAND;
    int r = flat % NCAND;
    const float* dat;
    int HW;
    unsigned a, hw, cand;
    if (r < NCAND0) {
        a = (unsigned)r / HW0; hw = (unsigned)r % HW0;
        dat = d0 + (size_t)b * CHN * HW0; HW = HW0;
        cand = hw * NA + a;
    } else {
        r -= NCAND0;
        a = (unsigned)r / HW1; hw = (unsigned)r % HW1;
        dat = d1 + (size_t)b * CHN * HW1; HW = HW1;
        cand = (unsigned)NCAND0 + hw * NA + a;
    }
    const float* sp = dat + (size_t)(a * (4 + NC) + 4) * HW + hw;
    float m = __builtin_nontemporal_load(sp);
#pragma unroll 8
    for (int c = 1; c < NC; ++c)
        m = fmaxf(m, __builtin_nontemporal_load(sp + (size_t)c * HW));
    scores[(size_t)b * NCAND + cand] = m;
}

// ---------------------------------------------------------------------------
// Kernel B: one 1024-thread (32-wave) block per image.
//   radix-select exact top-1000 -> bitonic sort desc -> gather+decode ->
//   greedy NMS (reference semantics) -> rank-preserving compaction.
// ---------------------------------------------------------------------------
__global__ __launch_bounds__(1024) void select_nms_kernel(
    const float* __restrict__ d0, const float* __restrict__ a0,
    const float* __restrict__ d1, const float* __restrict__ a1,
    const float* __restrict__ scores, float* __restrict__ out) {
    __shared__ unsigned sHist[256];
    __shared__ unsigned sMisc[4];
    __shared__ unsigned sKeys[1024];
    __shared__ unsigned sIdx[1024];
    __shared__ float    sTile[2][TILE];
    __shared__ float    sX1[1024], sY1[1024], sX2[1024], sY2[1024];
    __shared__ float    sSc[1024], sCl[1024];
    __shared__ unsigned sKeep[1024], sScan[1024];

    const int img = blockIdx.x;
    const unsigned tid = threadIdx.x;
    const float* sc = scores + (size_t)img * NCAND;

    float* oB = out;                              // [B,POST,4]
    float* oS = out + (size_t)NB * POST * 4;      // [B,POST]
    float* oC = oS + (size_t)NB * POST;           // [B,POST]

    const unsigned ldsT0 = (unsigned)(size_t)(&sTile[0][0]);
    const unsigned ldsT1 = (unsigned)(size_t)(&sTile[1][0]);

    // ---- 4-pass MSB radix select: T = key of the 1000th-largest score ----
    unsigned prefix = 0u, maskHi = 0u, need = PRE;
    for (int bp = 3; bp >= 0; --bp) {
        for (unsigned q = tid; q < 256u; q += 1024u) sHist[q] = 0u;
        __syncthreads();
        streamScores(sc, sTile, ldsT0, ldsT1, tid,
                     [&](unsigned e, float v) {
                         (void)e;
                         unsigned kk = keyOf(v);
                         if ((kk & maskHi) == prefix)
                             atomicAdd(&sHist[(kk >> (8 * bp)) & 255u], 1u);
                     });
        __syncthreads();
        if (tid == 0) {
            unsigned rem = need, chosen = 0u;
            for (int bin = 255; bin >= 0; --bin) {
                unsigned c = sHist[bin];
                if (rem <= c) { chosen = (unsigned)bin; break; }
                rem -= c;
            }
            sMisc[0] = chosen; sMisc[1] = rem;
        }
        __syncthreads();
        prefix |= sMisc[0] << (8 * bp);
        maskHi |= 0xFFu << (8 * bp);
        need = sMisc[1];
        __syncthreads();
    }
    const unsigned T = prefix;
    const unsigned G = PRE - need;   // exact count of keys strictly > T

    // ---- compaction: keys > T first, then `need` ties at T --------------
    sKeys[tid] = 0u; sIdx[tid] = 0xFFFFFFFFu;
    if (tid == 0) { sMisc[2] = 0u; sMisc[3] = 0u; }
    __syncthreads();
    streamScores(sc, sTile, ldsT0, ldsT1, tid,
                 [&](unsigned e, float v) {
                     unsigned kk = keyOf(v);
                     int slot = -1;
                     if (kk > T) {
                         slot = (int)atomicAdd(&sMisc[2], 1u);
                     } else if (kk == T) {
                         unsigned q = atomicAdd(&sMisc[3], 1u);
                         if (q < need) slot = (int)(G + q);
                     }
                     if (slot >= 0 && slot < PRE) { sKeys[slot] = kk; sIdx[slot] = e; }
                 });
    __syncthreads();

    // ---- bitonic sort, descending by key (1024 incl. zero-key pads) -----
    for (unsigned ksz = 2u; ksz <= 1024u; ksz <<= 1) {
        for (unsigned jj = ksz >> 1; jj > 0u; jj >>= 1) {
            __syncthreads();
            unsigned p = tid ^ jj;
            if (p > tid) {
                unsigned ka = sKeys[tid], kb = sKeys[p];
                bool desc = ((tid & ksz) == 0u);
                if ((ka < kb) == desc) {
                    sKeys[tid] = kb; sKeys[p] = ka;
                    unsigned ia = sIdx[tid]; sIdx[tid] = sIdx[p]; sIdx[p] = ia;
                }
            }
        }
    }
    __syncthreads();

    // ---- gather + decode the 1000 winners (reference BoxCoder + clip) ---
    {
        float score = 0.f, cls = 0.f, x1 = 0.f, y1 = 0.f, x2 = 0.f, y2 = 0.f;
        unsigned kp = 0u;
        unsigned ci = sIdx[tid];
        if (tid < PRE && ci != 0xFFFFFFFFu) {
            score = keyToFloat(sKeys[tid]);
            const float* dat; const float* anc; int HW; unsigned r = ci;
            if (r < NCAND0) {
                dat = d0 + (size_t)img * CHN * HW0;
                anc = a0 + (size_t)img * (NA * 4) * HW0; HW = HW0;
            } else {
                r -= NCAND0;
                dat = d1 + (size_t)img * CHN * HW1;
                anc = a1 + (size_t)img * (NA * 4) * HW1; HW = HW1;
            }
            unsigned hw = r / NA, a = r % NA;
            const float* dpt = dat + (size_t)(a * (4 + NC)) * HW + hw;
            float dx = dpt[0];
            float dy = dpt[(size_t)1 * HW];
            float dw = dpt[(size_t)2 * HW];
            float dh = dpt[(size_t)3 * HW];
            float best = dpt[(size_t)4 * HW]; int bi = 0;
            for (int c = 1; c < NC; ++c) {               // first-max == argmax
                float s = dpt[(size_t)(4 + c) * HW];
                if (s > best) { best = s; bi = c; }
            }
            cls = (float)bi;                             // CLASS_OFFSETS == 0
            const float* apt = anc + (size_t)(a * 4) * HW + hw;
            float ax1 = apt[0];
            float ay1 = apt[(size_t)1 * HW];
            float ax2 = apt[(size_t)2 * HW];
            float ay2 = apt[(size_t)3 * HW];
            float aw = ax2 - ax1, ah = ay2 - ay1;
            float acx = ax1 + 0.5f * aw, acy = ay1 + 0.5f * ah;
            float pcx = dx * aw + acx, pcy = dy * ah + acy;
            float pw = expf(fminf(dw, BBOX_CLIP)) * aw;
            float ph = expf(fminf(dh, BBOX_CLIP)) * ah;
            x1 = fminf(fmaxf(pcx - 0.5f * pw, 0.f), IMG_WH);
            y1 = fminf(fmaxf(pcy - 0.5f * ph, 0.f), IMG_WH);
            x2 = fminf(fmaxf(pcx + 0.5f * pw, 0.f), IMG_WH);
            y2 = fminf(fmaxf(pcy + 0.5f * ph, 0.f), IMG_WH);
            kp = (score >= SCORE_THR && x1 < x2 && y1 < y2) ? 1u : 0u;
        }
        sX1[tid] = x1; sY1[tid] = y1; sX2[tid] = x2; sY2[tid] = y2;
        sSc[tid] = score; sCl[tid] = cls; sKeep[tid] = kp;
    }
    __syncthreads();

    // ---- greedy NMS, exact reference semantics --------------------------
    for (int i = 0; i < PRE; ++i) {
        __syncthreads();
        if (sKeep[i] == 0u) continue;           // uniform branch
        unsigned j = tid;
        if (j > (unsigned)i && j < PRE && sKeep[j]) {
            float xx1 = fmaxf(sX1[i], sX1[j]);
            float yy1 = fmaxf(sY1[i], sY1[j]);
            float xx2 = fminf(sX2[i], sX2[j]);
            float yy2 = fminf(sY2[i], sY2[j]);
            float iw = fmaxf(xx2 - xx1, 0.f);
            float ih = fmaxf(yy2 - yy1, 0.f);
            float inter = iw * ih;
            float ai = (sX2[i] - sX1[i]) * (sY2[i] - sY1[i]);
            float aj = (sX2[j] - sX1[j]) * (sY2[j] - sY1[j]);
            float iou = inter / (ai + aj - inter + 1e-9f);
            if (iou > NMS_THR && (sSc[i] - sSc[j]) > 0.0f) sKeep[j] = 0u;
        }
    }
    __syncthreads();

    // ---- rank-preserving compaction -------------------------------------
    sScan[tid] = (tid < PRE) ? sKeep[tid] : 0u;
    __syncthreads();
    for (unsigned off = 1u; off < 1024u; off <<= 1) {
        unsigned add = (tid >= off) ? sScan[tid - off] : 0u;
        __syncthreads();
        sScan[tid] += add;
        __syncthreads();
    }
    // Build inverse map pos -> source slot in LDS (sIdx is dead after the
    // decode phase), so every output element is written EXACTLY once below —
    // no zero-then-overwrite WAW hazard between waves on global memory.
    if (tid < POST) sIdx[tid] = 0xFFFFFFFFu;
    __syncthreads();
    if (tid < PRE && sKeep[tid]) {
        unsigned pos = sScan[tid] - 1u;         // inclusive scan -> rank
        if (pos < POST) sIdx[pos] = tid;
    }
    __syncthreads();
    if (tid < POST) {
        unsigned j = sIdx[tid];
        size_t ob = ((size_t)img * POST + tid) * 4;
        if (j != 0xFFFFFFFFu) {
            oB[ob + 0] = sX1[j];
            oB[ob + 1] = sY1[j];
            oB[ob + 2] = sX2[j];
            oB[ob + 3] = sY2[j];
            oS[img * POST + tid] = sSc[j];
            oC[img * POST + tid] = sCl[j];
        } else {
            oB[ob + 0] = 0.0f; oB[ob + 1] = 0.0f;
            oB[ob + 2] = 0.0f; oB[ob + 3] = 0.0f;
            oS[img * POST + tid] = 0.0f;
            oC[img * POST + tid] = 0.0f;
        }
    }
}

// ---------------------------------------------------------------------------
extern "C" void kernel_launch(void* const* d_in, const int* in_sizes, int n_in,
                              void* d_out, int out_size, void* d_ws, size_t ws_size,
                              hipStream_t stream) {
    (void)in_sizes; (void)n_in; (void)out_size; (void)ws_size;
    // setup_inputs() order: data0, anchors0, data1, anchors1
    const float* d0 = (const float*)d_in[0];
    const float* a0 = (const float*)d_in[1];
    const float* d1 = (const float*)d_in[2];
    const float* a1 = (const float*)d_in[3];
    float* scores = (float*)d_ws;                 // NB*NCAND floats (2.3 MB)
    float* out = (float*)d_out;

    const int total = NB * NCAND;                 // 576000
    max_score_kernel<<<(total + 255) / 256, 256, 0, stream>>>(d0, d1, scores);
    select_nms_kernel<<<NB, 1024, 0, stream>>>(d0, a0, d1, a1, scores, out);
}